// EdgeGATLayer_72584947302718
// MI455X (gfx1250) — compile-verified
//
#include <hip/hip_runtime.h>
#include <hip/hip_bf16.h>

typedef __attribute__((ext_vector_type(2))) float v2f;
typedef __attribute__((ext_vector_type(8))) float v8f;

#define GAT_H 4
#define GAT_HD 32

// ---------- order-preserving float<->uint key for atomicMax on floats ----------
__device__ __forceinline__ unsigned fkey(float x) {
    unsigned u = __float_as_uint(x);
    return (u & 0x80000000u) ? ~u : (u | 0x80000000u);
}
__device__ __forceinline__ float fdec(unsigned k) {
    return (k & 0x80000000u) ? __uint_as_float(k & 0x7fffffffu) : __uint_as_float(~k);
}

// ---------- init: zero atomically-updated buffers + output ----------
__global__ void k_init(unsigned* __restrict__ emaxk, float* __restrict__ esum,
                       float* __restrict__ out, int ndst) {
    int stride = gridDim.x * blockDim.x;
    int i = blockIdx.x * blockDim.x + threadIdx.x;
    int n4 = ndst * 4, n128 = ndst * 128;
    for (int j = i; j < n4; j += stride) { emaxk[j] = 0u; esum[j] = 0.0f; }
    for (int j = i; j < n128; j += stride) out[j] = 0.0f;
}

// ---------- fold attention vectors into weights: B[k][h] = sum_d W[k][h*32+d]*a[h][d] ----------
__global__ void k_fold(const float* __restrict__ Wd, const float* __restrict__ We,
                       const float* __restrict__ ad, const float* __restrict__ ae,
                       float* __restrict__ Bd, float* __restrict__ Be) {
    int t = threadIdx.x;  // 768 threads: 512 for B_dst (128x4), 256 for B_edge (64x4)
    if (t < 512) {
        int k = t >> 2, h = t & 3;
        float s = 0.0f;
        #pragma unroll
        for (int d = 0; d < GAT_HD; d++) s += Wd[k * 128 + h * GAT_HD + d] * ad[h * GAT_HD + d];
        Bd[t] = s;
    } else {
        int t2 = t - 512;
        int k = t2 >> 2, h = t2 & 3;
        float s = 0.0f;
        #pragma unroll
        for (int d = 0; d < GAT_HD; d++) s += We[k * 128 + h * GAT_HD + d] * ae[h * GAT_HD + d];
        Be[t2] = s;
    }
}

// ---------- hs = h_src @ W_src via V_WMMA_F32_16X16X4_F32, one wave per 16x16 tile ----------
__global__ void k_gemm_hs(const float* __restrict__ A, const float* __restrict__ B,
                          float* __restrict__ C, int M) {
    const int K = 128, N = 128;
    int wave = (blockIdx.x * blockDim.x + threadIdx.x) >> 5;
    int lane = threadIdx.x & 31;
    const int tiles_n = N / 16;
    int tm = wave / tiles_n;
    int tn = wave - tm * tiles_n;
    if (tm * 16 >= M) return;

    int lh = lane >> 4;      // half-wave select (K pair for A/B, row-half for C/D)
    int ll = lane & 15;      // row (A) / col (B,C,D)
    int arow = tm * 16 + ll;
    if (arow >= M) arow = M - 1;          // clamp loads; stores are predicated below
    const float* Arow = A + (long long)arow * K;
    int bcol = tn * 16 + ll;

    v8f acc = {};
    #pragma unroll
    for (int k0 = 0; k0 < K; k0 += 4) {
        int ka = k0 + 2 * lh;            // lanes 16-31 hold K=+2,+3
        v2f a, b;
        a.x = Arow[ka];
        a.y = Arow[ka + 1];
        b.x = B[ka * N + bcol];
        b.y = B[(ka + 1) * N + bcol];
        acc = __builtin_amdgcn_wmma_f32_16x16x4_f32(
            /*neg_a=*/false, a, /*neg_b=*/false, b,
            /*c_mod=*/(short)0, acc, /*reuse_a=*/false, /*reuse_b=*/false);
    }
    #pragma unroll
    for (int i = 0; i < 8; i++) {
        int row = tm * 16 + i + 8 * lh;  // VGPR i: M=i (lanes 0-15), M=i+8 (lanes 16-31)
        if (row < M) C[(long long)row * N + tn * 16 + ll] = acc[i];
    }
}

// ---------- per-node logits: e_src_n from hs, e_dst_n = h_dst @ B_dst. One wave per node ----------
__global__ void k_node_logits(const float* __restrict__ hs, const float* __restrict__ hdst,
                              const float* __restrict__ a_src, const float* __restrict__ Bdst,
                              float* __restrict__ esrc, float* __restrict__ edst,
                              int nsrc, int ndst) {
    int wave = (blockIdx.x * blockDim.x + threadIdx.x) >> 5;
    int lane = threadIdx.x & 31;
    int h = lane >> 3;       // head owned by this lane's column range
    int c0 = lane * 4;       // column base (0..124)

    if (wave < nsrc) {
        const float4 v = *(const float4*)(hs + (long long)wave * 128 + c0);
        int d0 = c0 & 31;
        float s = v.x * a_src[h * GAT_HD + d0]     + v.y * a_src[h * GAT_HD + d0 + 1]
                + v.z * a_src[h * GAT_HD + d0 + 2] + v.w * a_src[h * GAT_HD + d0 + 3];
        // reduce within each group of 8 lanes (one head per group)
        s += __shfl_xor(s, 1, 8);
        s += __shfl_xor(s, 2, 8);
        s += __shfl_xor(s, 4, 8);
        if ((lane & 7) == 0) esrc[wave * 4 + h] = s;
    }
    if (wave < ndst) {
        const float4 v = *(const float4*)(hdst + (long long)wave * 128 + c0);
        float acc[4];
        #pragma unroll
        for (int hh = 0; hh < 4; hh++) {
            acc[hh] = v.x * Bdst[(c0    ) * 4 + hh] + v.y * Bdst[(c0 + 1) * 4 + hh]
                    + v.z * Bdst[(c0 + 2) * 4 + hh] + v.w * Bdst[(c0 + 3) * 4 + hh];
        }
        #pragma unroll
        for (int off = 16; off >= 1; off >>= 1) {
            #pragma unroll
            for (int hh = 0; hh < 4; hh++) acc[hh] += __shfl_down(acc[hh], off, 32);
        }
        if (lane == 0) {
            #pragma unroll
            for (int hh = 0; hh < 4; hh++) edst[wave * 4 + hh] = acc[hh];
        }
    }
}

// ---------- per-edge logits + leaky-relu + atomic max over destination segments ----------
__global__ void k_edge_logits(const float* __restrict__ ef, const int* __restrict__ src,
                              const int* __restrict__ dst, const float* __restrict__ Bedge,
                              const float* __restrict__ esrc, const float* __restrict__ edst,
                              float* __restrict__ e_ws, unsigned* __restrict__ emaxk, int E) {
    int e = blockIdx.x * blockDim.x + threadIdx.x;
    if (e >= E) return;
    const float* row = ef + (long long)e * 64;
    float a0 = 0.f, a1 = 0.f, a2 = 0.f, a3 = 0.f;
    #pragma unroll
    for (int k = 0; k < 64; k++) {
        float v = row[k];
        const float4 bw = *(const float4*)(Bedge + k * 4);
        a0 += v * bw.x; a1 += v * bw.y; a2 += v * bw.z; a3 += v * bw.w;
    }
    int s = src[e], d = dst[e];
    float eh[4] = {a0, a1, a2, a3};
    #pragma unroll
    for (int h = 0; h < 4; h++) {
        float t = eh[h] + esrc[s * 4 + h] + edst[d * 4 + h];
        t = t > 0.0f ? t : 0.2f * t;             // leaky_relu(0.2)
        e_ws[(long long)e * 4 + h] = t;
        atomicMax(&emaxk[d * 4 + h], fkey(t));
    }
}

// ---------- exp(e - max) and atomic segment sum ----------
__global__ void k_exp_sum(const int* __restrict__ dst, const unsigned* __restrict__ emaxk,
                          float* __restrict__ e_ws, float* __restrict__ esum, int E) {
    int e = blockIdx.x * blockDim.x + threadIdx.x;
    if (e >= E) return;
    int d = dst[e];
    #pragma unroll
    for (int h = 0; h < 4; h++) {
        float m = fdec(emaxk[d * 4 + h]);
        float ex = expf(e_ws[(long long)e * 4 + h] - m);
        e_ws[(long long)e * 4 + h] = ex;
        atomicAdd(&esum[d * 4 + h], ex);
    }
}

// ---------- weighted scatter: out[dst] += alpha * hs[src]. One wave per edge ----------
__global__ void k_scatter(const float* __restrict__ hs, const int* __restrict__ src,
                          const int* __restrict__ dst, const float* __restrict__ e_ws,
                          const float* __restrict__ esum, float* __restrict__ out, int E) {
    int wave = (blockIdx.x * blockDim.x + threadIdx.x) >> 5;
    int lane = threadIdx.x & 31;
    if (wave >= E) return;
    int s = src[wave], d = dst[wave];
    int h = lane >> 3;
    float alpha = e_ws[(long long)wave * 4 + h] / (esum[d * 4 + h] + 1e-8f);
    const float4 v = *(const float4*)(hs + (long long)s * 128 + lane * 4);
    float* o = out + (long long)d * 128 + lane * 4;
    atomicAdd(o + 0, alpha * v.x);
    atomicAdd(o + 1, alpha * v.y);
    atomicAdd(o + 2, alpha * v.z);
    atomicAdd(o + 3, alpha * v.w);
}

extern "C" void kernel_launch(void* const* d_in, const int* in_sizes, int n_in,
                              void* d_out, int out_size, void* d_ws, size_t ws_size,
                              hipStream_t stream) {
    const float* h_src     = (const float*)d_in[0];
    const float* h_dst     = (const float*)d_in[1];
    const float* edge_feat = (const float*)d_in[2];
    const int*   edge_idx  = (const int*)  d_in[3];
    const float* W_src     = (const float*)d_in[4];
    const float* W_dst     = (const float*)d_in[5];
    const float* W_edge    = (const float*)d_in[6];
    const float* a_src     = (const float*)d_in[7];
    const float* a_dst     = (const float*)d_in[8];
    const float* a_edge    = (const float*)d_in[9];

    int nsrc = in_sizes[0] / 128;
    int ndst = in_sizes[1] / 128;
    int E    = in_sizes[2] / 64;
    const int* src = edge_idx;
    const int* dst = edge_idx + E;

    float* out = (float*)d_out;

    // workspace layout
    float*    ws    = (float*)d_ws;
    float*    hs    = ws;                                   // nsrc*128
    float*    e_ws  = hs    + (size_t)nsrc * 128;           // E*4
    float*    esrc  = e_ws  + (size_t)E * 4;                // nsrc*4
    float*    edstb = esrc  + (size_t)nsrc * 4;             // ndst*4
    unsigned* emaxk = (unsigned*)(edstb + (size_t)ndst * 4);// ndst*4
    float*    esum  = (float*)(emaxk + (size_t)ndst * 4);   // ndst*4
    float*    Bdst  = esum  + (size_t)ndst * 4;             // 512
    float*    Bedge = Bdst  + 512;                          // 256

    // 1. init atomically-updated buffers and output
    k_init<<<1024, 256, 0, stream>>>(emaxk, esum, out, ndst);

    // 2. fold a_dst/a_edge into weight matrices (tiny)
    k_fold<<<1, 768, 0, stream>>>(W_dst, W_edge, a_dst, a_edge, Bdst, Bedge);

    // 3. hs = h_src @ W_src (fp32 WMMA), one wave per 16x16 tile
    int tiles_m = (nsrc + 15) / 16;
    int waves_g = tiles_m * (128 / 16);
    k_gemm_hs<<<(waves_g + 7) / 8, 256, 0, stream>>>(h_src, W_src, hs, nsrc);

    // 4. per-node attention logit components
    int nmax = nsrc > ndst ? nsrc : ndst;
    k_node_logits<<<(nmax + 7) / 8, 256, 0, stream>>>(hs, h_dst, a_src, Bdst,
                                                      esrc, edstb, nsrc, ndst);

    // 5. per-edge logits + segment max
    k_edge_logits<<<(E + 255) / 256, 256, 0, stream>>>(edge_feat, src, dst, Bedge,
                                                       esrc, edstb, e_ws, emaxk, E);

    // 6. exp + segment sum
    k_exp_sum<<<(E + 255) / 256, 256, 0, stream>>>(dst, emaxk, e_ws, esum, E);

    // 7. alpha-weighted scatter of hs[src] into out[dst]
    k_scatter<<<(E + 7) / 8, 256, 0, stream>>>(hs, src, dst, e_ws, esum, out, E);
}